// EnhancedGNN_27273042329839
// MI455X (gfx1250) — compile-verified
//
#include <hip/hip_runtime.h>
#include <hip/hip_bf16.h>
#include <math.h>

typedef __attribute__((ext_vector_type(2))) float v2f;
typedef __attribute__((ext_vector_type(8))) float v8f;

#define N_NODES 2048
#define F_IN    128
#define H_DIM   16
#define N_PAIRS (N_NODES * N_NODES)

// -------------------------------------------------------------------------
// WMMA fp32 GEMM:  OUT[M,16] = IN[M,K] @ W[K,16]   (exact fp32, 16x16x4)
// One wave32 per 16-row tile of M. K must be a multiple of 4.
// A 16x4 layout (ISA 7.12.2): lanes 0-15 -> M=lane, VGPR0=K0 VGPR1=K1;
//                             lanes 16-31 -> M=lane-16, VGPR0=K2 VGPR1=K3.
// C/D 16x16:  VGPR r: lanes 0-15 -> (M=r, N=lane); lanes 16-31 -> (M=r+8).
// -------------------------------------------------------------------------
__global__ void wmma_gemm_n16(const float* __restrict__ in,
                              const float* __restrict__ w,
                              float* __restrict__ out, int K) {
  const int wave = (blockIdx.x * blockDim.x + threadIdx.x) >> 5;
  const int lane = threadIdx.x & 31;
  const int half = lane >> 4;      // 0 -> K{0,1}, 1 -> K{2,3}
  const int l    = lane & 15;
  const int row0 = wave * 16;

  v8f acc = {};
  for (int k = 0; k < K; k += 4) {
    const int ka = k + half * 2;
    v2f a, b;
    a.x = in[(row0 + l) * K + ka + 0];
    a.y = in[(row0 + l) * K + ka + 1];
    b.x = w[(ka + 0) * 16 + l];
    b.y = w[(ka + 1) * 16 + l];
    // 8 args: (neg_a, A, neg_b, B, c_mod, C, reuse_a, reuse_b)
    acc = __builtin_amdgcn_wmma_f32_16x16x4_f32(false, a, false, b,
                                                (short)0, acc, false, false);
  }
#pragma unroll
  for (int r = 0; r < 8; ++r) {
    out[(row0 + r + half * 8) * 16 + l] = acc[r];
  }
}

// ------------------------- degree / normalization -------------------------
__global__ void deg_init_kernel(float* __restrict__ deg) {
  int i = blockIdx.x * blockDim.x + threadIdx.x;
  if (i < N_NODES) deg[i] = 1.0f;   // self loop
}

__global__ void deg_scatter_kernel(const int* __restrict__ dst,
                                   float* __restrict__ deg, int E) {
  int e = blockIdx.x * blockDim.x + threadIdx.x;
  if (e < E) atomicAdd(&deg[dst[e]], 1.0f);
}

__global__ void dinv_kernel(float* __restrict__ deg) {
  int i = blockIdx.x * blockDim.x + threadIdx.x;
  if (i < N_NODES) deg[i] = rsqrtf(deg[i]);   // deg >= 1 (self loop)
}

// ------------------------- GCN conv helpers -------------------------------
// acc[i,f] = dinv[i]^2 * xw[i,f]   (self-loop message, norm = dinv_i*dinv_i)
__global__ void conv_self_init_kernel(const float* __restrict__ xw,
                                      const float* __restrict__ dinv,
                                      float* __restrict__ acc, int F) {
  int t = blockIdx.x * blockDim.x + threadIdx.x;
  if (t < N_NODES * F) {
    int i = t / F;
    float di = dinv[i];
    acc[t] = di * di * xw[t];
  }
}

__global__ void conv_scatter_kernel(const int* __restrict__ src,
                                    const int* __restrict__ dst,
                                    const float* __restrict__ xw,
                                    const float* __restrict__ dinv,
                                    float* __restrict__ acc, int E, int F) {
  int e = blockIdx.x * blockDim.x + threadIdx.x;
  if (e >= E) return;
  int s = src[e], d = dst[e];
  float nrm = dinv[s] * dinv[d];
  for (int f = 0; f < F; ++f)
    atomicAdd(&acc[d * F + f], xw[s * F + f] * nrm);
}

// h = relu(acc + b)   (in place)
__global__ void bias_relu_kernel(float* __restrict__ acc,
                                 const float* __restrict__ b, int F) {
  int t = blockIdx.x * blockDim.x + threadIdx.x;
  if (t < N_NODES * F) {
    float v = acc[t] + b[t % F];
    acc[t] = fmaxf(v, 0.0f);
  }
}

// out[i,c] = acc[i,c] + b[c]   written to d_out (node_out, [2048,2])
__global__ void node_out_kernel(const float* __restrict__ acc,
                                const float* __restrict__ b,
                                float* __restrict__ out) {
  int t = blockIdx.x * blockDim.x + threadIdx.x;
  if (t < N_NODES * 2) out[t] = acc[t] + b[t & 1];
}

// xw2[i,c] = sum_f h[i,f] * W2[f,c]   (N=2: plain VALU, WMMA pointless)
__global__ void gemv2_kernel(const float* __restrict__ h,
                             const float* __restrict__ W2,
                             float* __restrict__ xw2) {
  int i = blockIdx.x * blockDim.x + threadIdx.x;
  if (i >= N_NODES) return;
  float a0 = 0.f, a1 = 0.f;
#pragma unroll
  for (int f = 0; f < H_DIM; ++f) {
    float hv = h[i * H_DIM + f];
    a0 = fmaf(hv, W2[f * 2 + 0], a0);
    a1 = fmaf(hv, W2[f * 2 + 1], a1);
  }
  xw2[i * 2 + 0] = a0;
  xw2[i * 2 + 1] = a1;
}

// ------------------------- all-pairs edge MLP -----------------------------
// edge_out[i*N+j] = sigmoid( relu(A[i,:]+B[j,:]+bc1) . Wc2 + bc2 )
// 16x16 tile per 256-thread block; A/B tiles staged through LDS.
__global__ void edge_mlp_kernel(const float* __restrict__ A,
                                const float* __restrict__ B,
                                const float* __restrict__ Wc2,
                                const float* __restrict__ bc1,
                                const float* __restrict__ bc2,
                                float* __restrict__ edge_out,
                                float* __restrict__ start_out,
                                float* __restrict__ end_out) {
  __shared__ float sA[16][17];
  __shared__ float sB[16][17];
  __shared__ float sW[16];
  __shared__ float sb1[16];

  const int tx    = threadIdx.x;       // 256 threads
  const int j_loc = tx & 15;
  const int i_loc = tx >> 4;
  const int i0 = blockIdx.y * 16;
  const int j0 = blockIdx.x * 16;

  sA[i_loc][j_loc] = A[(i0 + i_loc) * H_DIM + j_loc];
  sB[i_loc][j_loc] = B[(j0 + i_loc) * H_DIM + j_loc];
  if (tx < 16) { sW[tx] = Wc2[tx]; sb1[tx] = bc1[tx]; }
  __syncthreads();

  const int i = i0 + i_loc;
  const int j = j0 + j_loc;
  float s = 0.f;
#pragma unroll
  for (int f = 0; f < H_DIM; ++f) {
    float hv = sA[i_loc][f] + sB[j_loc][f] + sb1[f];
    hv = fmaxf(hv, 0.0f);
    s = fmaf(hv, sW[f], s);
  }
  s += bc2[0];
  float sig = 1.0f / (1.0f + __expf(-s));

  int p = i * N_NODES + j;             // < 2^31
  edge_out[p]  = sig;
  start_out[p] = (float)i;
  end_out[p]   = (float)j;
}

// -------------------------------------------------------------------------
extern "C" void kernel_launch(void* const* d_in, const int* in_sizes, int n_in,
                              void* d_out, int out_size, void* d_ws, size_t ws_size,
                              hipStream_t stream) {
  const float* x    = (const float*)d_in[0];
  const int*   eidx = (const int*)  d_in[1];   // [2, E]
  const float* W1   = (const float*)d_in[2];
  const float* b1   = (const float*)d_in[3];
  const float* W2   = (const float*)d_in[4];
  const float* b2   = (const float*)d_in[5];
  const float* Wc1  = (const float*)d_in[6];   // [32,16] row-major
  const float* bc1  = (const float*)d_in[7];
  const float* Wc2  = (const float*)d_in[8];   // [16,1]
  const float* bc2  = (const float*)d_in[9];

  const int E = in_sizes[1] / 2;
  const int* src = eidx;
  const int* dst = eidx + E;

  // workspace layout (floats)
  float* ws    = (float*)d_ws;
  float* xw1   = ws;                    // 2048*16
  float* hbuf  = xw1  + N_NODES * H_DIM; // 2048*16 (acc -> h in place)
  float* dinv  = hbuf + N_NODES * H_DIM; // 2048 (deg -> dinv in place)
  float* Abuf  = dinv + N_NODES;        // 2048*16
  float* Bbuf  = Abuf + N_NODES * H_DIM;// 2048*16
  float* xw2   = Bbuf + N_NODES * H_DIM;// 2048*2
  float* nacc  = xw2  + N_NODES * 2;    // 2048*2

  // output layout: node_out [2048*2] | edge_out [N^2] | start [N^2] | end [N^2]
  float* out       = (float*)d_out;
  float* node_out  = out;
  float* edge_out  = out + N_NODES * 2;
  float* start_out = edge_out + N_PAIRS;
  float* end_out   = start_out + N_PAIRS;

  // --- degree / symmetric normalization ---
  deg_init_kernel<<<(N_NODES + 255) / 256, 256, 0, stream>>>(dinv);
  deg_scatter_kernel<<<(E + 255) / 256, 256, 0, stream>>>(dst, dinv, E);
  dinv_kernel<<<(N_NODES + 255) / 256, 256, 0, stream>>>(dinv);

  // --- layer 1: xw1 = x @ W1 (WMMA fp32), then normalized aggregation ---
  {
    int waves = N_NODES / 16;                 // 128
    wmma_gemm_n16<<<(waves * 32) / 256, 256, 0, stream>>>(x, W1, xw1, F_IN);
  }
  conv_self_init_kernel<<<(N_NODES * H_DIM + 255) / 256, 256, 0, stream>>>(xw1, dinv, hbuf, H_DIM);
  conv_scatter_kernel<<<(E + 255) / 256, 256, 0, stream>>>(src, dst, xw1, dinv, hbuf, E, H_DIM);
  bias_relu_kernel<<<(N_NODES * H_DIM + 255) / 256, 256, 0, stream>>>(hbuf, b1, H_DIM);

  // --- edge-MLP factorization: A = h @ Wc1[0:16,:], B = h @ Wc1[16:32,:] ---
  {
    int waves = N_NODES / 16;
    wmma_gemm_n16<<<(waves * 32) / 256, 256, 0, stream>>>(hbuf, Wc1, Abuf, H_DIM);
    wmma_gemm_n16<<<(waves * 32) / 256, 256, 0, stream>>>(hbuf, Wc1 + 16 * H_DIM, Bbuf, H_DIM);
  }

  // --- layer 2: xw2 = h @ W2, aggregate, node_out = acc + b2 ---
  gemv2_kernel<<<(N_NODES + 255) / 256, 256, 0, stream>>>(hbuf, W2, xw2);
  conv_self_init_kernel<<<(N_NODES * 2 + 255) / 256, 256, 0, stream>>>(xw2, dinv, nacc, 2);
  conv_scatter_kernel<<<(E + 255) / 256, 256, 0, stream>>>(src, dst, xw2, dinv, nacc, E, 2);
  node_out_kernel<<<(N_NODES * 2 + 255) / 256, 256, 0, stream>>>(nacc, b2, node_out);

  // --- all-pairs edge MLP (store-bandwidth bound; 16x16 LDS tiles) ---
  {
    dim3 grid(N_NODES / 16, N_NODES / 16);
    edge_mlp_kernel<<<grid, 256, 0, stream>>>(Abuf, Bbuf, Wc2, bc1, bc2,
                                              edge_out, start_out, end_out);
  }
}